// TNorm_71038759076547
// MI455X (gfx1250) — compile-verified
//
#include <hip/hip_runtime.h>

typedef __attribute__((ext_vector_type(2))) float v2f;
typedef __attribute__((ext_vector_type(8))) float v8f;

// out[b, ((i*8+j)*8+k)*8+l] = g0[i]*g1[j]*g2[k]*g3[l]
//   = p[i*8+j] * q[k*8+l]  with p = kron(g0,g1), q = kron(g2,g3)
// i.e. per batch row a 64x64 rank-1 outer product, row-major flattened.
// Tile as 4x4 grid of 16x16 outer products, each one V_WMMA_F32_16X16X4_F32
// with the vectors placed in A column K=0 / B row K=0 (K=1..3 zero).
__global__ __launch_bounds__(256) void tnorm_outer_kernel(
    const float* __restrict__ x, float* __restrict__ out, int batch) {
  const int lane = threadIdx.x & 31;
  const int wave = blockIdx.x * (blockDim.x >> 5) + (threadIdx.x >> 5);
  if (wave >= batch) return;  // wave-uniform: EXEC stays all-1s for live waves

  const float* __restrict__ xb = x + (size_t)wave * 32;
  float* __restrict__ ob = out + (size_t)wave * 4096;

  const int  L  = lane & 15;
  const bool lo = lane < 16;

  // Lane L (0..15) needs p[t*16+L] and q[t*16+L] for the 4 tile rows/cols.
  // p[m] = g0[m>>3]*g1[m&7], q[n] = g2[n>>3]*g3[n&7].
  float pa[4], qb[4];
#pragma unroll
  for (int t = 0; t < 4; ++t) {
    const int idx = t * 16 + L;
    const float g0 = xb[idx >> 3];
    const float g1 = xb[8 + (idx & 7)];
    const float g2 = xb[16 + (idx >> 3)];
    const float g3 = xb[24 + (idx & 7)];
    pa[t] = lo ? (g0 * g1) : 0.0f;  // lanes 16..31 hold K=2 slice -> zero
    qb[t] = lo ? (g2 * g3) : 0.0f;
  }

  const int rowAdd = lo ? 0 : 8;  // D layout: VGPR r -> M=r (lo) / M=r+8 (hi)

#pragma unroll
  for (int ti = 0; ti < 4; ++ti) {
    v2f a;
    a.x = pa[ti];   // A: K=0 column = p chunk (lanes 0-15), else 0
    a.y = 0.0f;     // K=1 (lo) / K=3 (hi) slice = 0
#pragma unroll
    for (int tj = 0; tj < 4; ++tj) {
      v2f b;
      b.x = qb[tj]; // B: K=0 row = q chunk (lanes 0-15), else 0
      b.y = 0.0f;
      v8f c = {};
      // D[M,N] = p[ti*16+M] * q[tj*16+N]
      v8f d = __builtin_amdgcn_wmma_f32_16x16x4_f32(
          /*neg_a=*/false, a, /*neg_b=*/false, b,
          /*c_mod=*/(short)0, c, /*reuse_a=*/false, /*reuse_b=*/false);

      float* tbase = ob + (size_t)(ti * 16 + rowAdd) * 64 + tj * 16 + L;
#pragma unroll
      for (int r = 0; r < 8; ++r) {
        // row-major: VGPR r across lanes = one contiguous 64B row segment.
        // Output (256 MB) is write-once, > L2 (192 MB): use non-temporal.
        __builtin_nontemporal_store(d[r], tbase + (size_t)r * 64);
      }
    }
  }
}

extern "C" void kernel_launch(void* const* d_in, const int* in_sizes, int n_in,
                              void* d_out, int out_size, void* d_ws, size_t ws_size,
                              hipStream_t stream) {
  const float* x = (const float*)d_in[0];
  float* out = (float*)d_out;
  const int batch = in_sizes[0] / 32;          // 16384 rows of 4*8 floats
  const int waves_per_block = 256 / 32;        // wave32: 8 waves / block
  const int blocks = (batch + waves_per_block - 1) / waves_per_block;
  tnorm_outer_kernel<<<blocks, 256, 0, stream>>>(x, out, batch);
}